// AlternateCrossAttentionBlock_4724464026144
// MI455X (gfx1250) — compile-verified
//
#include <hip/hip_runtime.h>
#include <math.h>

// ---------------------------------------------------------------------------
// Problem constants (from reference)
// ---------------------------------------------------------------------------
constexpr int Bn  = 4;     // batch
constexpr int Wd  = 32;    // width
constexpr int Hd  = 32;    // height
constexpr int Cc  = 128;   // image channels
constexpr int Ss  = 256;   // text seq len
constexpr int Dd  = 512;   // text embed dim
constexpr int NC  = 64;    // attention inner dim
constexpr int NPOS = Bn * Wd * Hd;  // 4096 spatial positions

typedef __attribute__((ext_vector_type(2))) float v2f;
typedef __attribute__((ext_vector_type(8))) float v8f;

// D = A(16x4 f32) * B(4x16 f32) + C(16x16 f32)   -- CDNA5 V_WMMA_F32_16X16X4_F32
__device__ __forceinline__ v8f wmma_k4(v2f a, v2f b, v8f c) {
  return __builtin_amdgcn_wmma_f32_16x16x4_f32(
      /*neg_a=*/false, a, /*neg_b=*/false, b,
      /*c_mod=*/(short)0, c, /*reuse_a=*/false, /*reuse_b=*/false);
}

// f32 WMMA fragment index helpers (ISA 7.12.2):
//  A 16x4 : lane L -> M = L&15 ; K pair = (L>>4)*2 .. +1  (vgpr0=K, vgpr1=K+1)
//  B 4x16 : lane L -> N = L&15 ; K pair = (L>>4)*2
//  C/D    : vgpr i, lane L -> M = i + 8*(L>>4), N = L&15

// ---------------------------------------------------------------------------
// Kernel 1: ks/vs = (s * a) @ W + b     per batch: (256x512)@(512x64)
// grid = B * (S/16) * 2 blocks, 128 threads (4 waves = 4 N-tiles of 16)
// ---------------------------------------------------------------------------
__global__ __launch_bounds__(128) void text_proj_kernel(
    const float* __restrict__ s, const float* __restrict__ amask,
    const float* __restrict__ Wks, const float* __restrict__ bks,
    const float* __restrict__ Wvs, const float* __restrict__ bvs,
    float* __restrict__ ks_out, float* __restrict__ vs_out) {
  constexpr int TE_STRIDE = Dd + 4;              // 516: stride%64==4 -> no bank conflicts
  __shared__ float te_s[16 * TE_STRIDE];         // 33 KB

  const int mat = blockIdx.x & 1;                // 0 -> ks, 1 -> vs
  const int st  = (blockIdx.x >> 1) & 15;        // S tile
  const int b   = blockIdx.x >> 5;               // batch

  const int tid  = threadIdx.x;
  const int lane = tid & 31;
  const int wave = tid >> 5;

  // stage te tile (16 rows x 512) into LDS, applying the text mask
  for (int i = tid; i < 16 * Dd; i += 128) {
    const int r = i >> 9, c = i & (Dd - 1);
    const int srow = st * 16 + r;
    te_s[r * TE_STRIDE + c] =
        s[((size_t)b * Ss + srow) * Dd + c] * amask[b * Ss + srow];
  }
  __syncthreads();

  const float* Wm   = mat ? Wvs : Wks;
  const float* bias = mat ? bvs : bks;
  float*       dst  = mat ? vs_out : ks_out;

  const int m    = lane & 15;
  const int koff = (lane >> 4) * 2;
  const int n    = wave * 16 + (lane & 15);

  v8f acc = {0.f, 0.f, 0.f, 0.f, 0.f, 0.f, 0.f, 0.f};
#pragma unroll 4
  for (int k = 0; k < Dd; k += 4) {
    v2f af, bf;
    af.x = te_s[m * TE_STRIDE + k + koff];
    af.y = te_s[m * TE_STRIDE + k + koff + 1];
    bf.x = Wm[(k + koff) * NC + n];
    bf.y = Wm[(k + koff + 1) * NC + n];
    acc = wmma_k4(af, bf, acc);
  }
  const float bn = bias[n];
#pragma unroll
  for (int i = 0; i < 8; ++i) {
    const int row = i + ((lane >> 4) << 3);
    dst[((size_t)b * Ss + st * 16 + row) * NC + n] = acc[i] + bn;
  }
}

// ---------------------------------------------------------------------------
// Kernel 2: Ksum[b,d] = sum_s ks ; Vsum[b,c] = sum_s vs   (tiny)
// grid = B, 128 threads
// ---------------------------------------------------------------------------
__global__ __launch_bounds__(128) void colsum_kernel(
    const float* __restrict__ ks, const float* __restrict__ vs,
    float* __restrict__ Ksum, float* __restrict__ Vsum) {
  const int b = blockIdx.x;
  const int t = threadIdx.x;
  const float* src = (t < NC) ? ks : vs;
  const int col = t & (NC - 1);
  float acc = 0.f;
  for (int sIdx = 0; sIdx < Ss; ++sIdx)
    acc += src[((size_t)b * Ss + sIdx) * NC + col];
  if (t < NC) Ksum[b * NC + col] = acc;
  else        Vsum[b * NC + col] = acc;
}

// ---------------------------------------------------------------------------
// Kernel 3: G[b] = vs[b]^T @ ks[b]  (64x64, K=256)
// grid = B * (NC/16) (M tiles), 128 threads (4 waves = 4 N-tiles)
// ---------------------------------------------------------------------------
__global__ __launch_bounds__(128) void gram_kernel(
    const float* __restrict__ ks, const float* __restrict__ vs,
    float* __restrict__ G) {
  const int mt = blockIdx.x & 3;   // C-row tile
  const int b  = blockIdx.x >> 2;
  const int lane = threadIdx.x & 31;
  const int wave = threadIdx.x >> 5;

  const int m    = lane & 15;
  const int koff = (lane >> 4) * 2;
  const int c0   = mt * 16;
  const int n    = wave * 16 + (lane & 15);

  v8f acc = {0.f, 0.f, 0.f, 0.f, 0.f, 0.f, 0.f, 0.f};
#pragma unroll 4
  for (int k = 0; k < Ss; k += 4) {
    v2f af, bf;
    // A[m][kk] = vs[b][kk][c0+m]  (vs^T)
    af.x = vs[((size_t)b * Ss + k + koff) * NC + c0 + m];
    af.y = vs[((size_t)b * Ss + k + koff + 1) * NC + c0 + m];
    bf.x = ks[((size_t)b * Ss + k + koff) * NC + n];
    bf.y = ks[((size_t)b * Ss + k + koff + 1) * NC + n];
    acc = wmma_k4(af, bf, acc);
  }
#pragma unroll
  for (int i = 0; i < 8; ++i) {
    const int row = i + ((lane >> 4) << 3);
    G[((size_t)b * NC + c0 + row) * NC + n] = acc[i];
  }
}

// ---------------------------------------------------------------------------
// Kernel 4 (fused): per 16 positions:
//   q/kx/vx = x@W+b (WMMA) -> attn = (S*vx*kx + vx*Ksum + kx*Vsum + G)/8
//   -> row softmax -> out = q@attn -> @Wout+bout -> residual -> LayerNorm
// grid = NPOS/16 = 256 blocks, 128 threads (4 waves)
// ---------------------------------------------------------------------------
__global__ __launch_bounds__(128) void fused_attn_kernel(
    const float* __restrict__ x,
    const float* __restrict__ Wq,  const float* __restrict__ bq,
    const float* __restrict__ Wkx, const float* __restrict__ bkx,
    const float* __restrict__ Wvx, const float* __restrict__ bvx,
    const float* __restrict__ G,
    const float* __restrict__ Ksum, const float* __restrict__ Vsum,
    const float* __restrict__ Wout, const float* __restrict__ bout,
    const float* __restrict__ ln_scale, const float* __restrict__ ln_bias,
    float* __restrict__ y) {
  constexpr int XS = Cc + 4;   // 132: %64==4 -> conflict-free A-frag reads
  constexpr int QS = NC + 4;   // 68
  __shared__ float xs[16 * XS];        // x tile (residual source + A operand)
  __shared__ float qs[16 * QS];
  __shared__ float kxs[16 * QS];
  __shared__ float vxs[16 * QS];
  __shared__ float os[16 * QS];        // attention output (A operand for Wout)
  __shared__ float outs[16 * XS];      // Wout result
  __shared__ float Gs[NC * NC];        // 16 KB
  __shared__ float KsS[NC], VsS[NC];

  const int tid  = threadIdx.x;
  const int lane = tid & 31;
  const int wave = tid >> 5;
  const int p0   = blockIdx.x * 16;          // first global position of tile
  const int b    = p0 >> 10;                 // W*H = 1024 positions per batch

  // ---- stage x tile, G, Ksum/Vsum ----
  for (int i = tid; i < 16 * Cc; i += 128)
    xs[(i >> 7) * XS + (i & (Cc - 1))] = x[(size_t)p0 * Cc + i];
  for (int i = tid; i < NC * NC; i += 128)
    Gs[i] = G[(size_t)b * NC * NC + i];
  if (tid < NC)            KsS[tid] = Ksum[b * NC + tid];
  else if (tid < 2 * NC)   VsS[tid - NC] = Vsum[b * NC + tid - NC];
  __syncthreads();

  // ---- phase 2: q/kx/vx projections, 16x64 each via WMMA (K=128) ----
  {
    const int m    = lane & 15;
    const int koff = (lane >> 4) * 2;
    const int n    = wave * 16 + (lane & 15);
    for (int mat = 0; mat < 3; ++mat) {
      const float* Wm   = (mat == 0) ? Wq  : (mat == 1) ? Wkx : Wvx;
      const float* bias = (mat == 0) ? bq  : (mat == 1) ? bkx : bvx;
      float*       dst  = (mat == 0) ? qs  : (mat == 1) ? kxs : vxs;
      v8f acc = {0.f, 0.f, 0.f, 0.f, 0.f, 0.f, 0.f, 0.f};
#pragma unroll 4
      for (int k = 0; k < Cc; k += 4) {
        v2f af, bf;
        af.x = xs[m * XS + k + koff];
        af.y = xs[m * XS + k + koff + 1];
        bf.x = Wm[(k + koff) * NC + n];
        bf.y = Wm[(k + koff + 1) * NC + n];
        acc = wmma_k4(af, bf, acc);
      }
      const float bn = bias[n];
#pragma unroll
      for (int i = 0; i < 8; ++i)
        dst[(i + ((lane >> 4) << 3)) * QS + n] = acc[i] + bn;
    }
  }
  __syncthreads();

  // ---- phase 3: per-position channel-channel attention (wave32 softmax) ----
  // attn[j][d] = (256*vx[j]*kx[d] + vx[j]*Ksum[d] + kx[d]*Vsum[j] + G[j][d])/8
  for (int pi = 0; pi < 4; ++pi) {
    const int p = wave * 4 + pi;
    const float kx0 = kxs[p * QS + lane];
    const float kx1 = kxs[p * QS + lane + 32];
    const float Ks0 = KsS[lane];
    const float Ks1 = KsS[lane + 32];
    float o0 = 0.f, o1 = 0.f;
#pragma unroll 4
    for (int j = 0; j < NC; ++j) {
      const float vj = vxs[p * QS + j];
      const float Vj = VsS[j];
      const float qj = qs[p * QS + j];
      float r0 = (256.f * vj * kx0 + vj * Ks0 + kx0 * Vj + Gs[j * NC + lane]) * 0.125f;
      float r1 = (256.f * vj * kx1 + vj * Ks1 + kx1 * Vj + Gs[j * NC + lane + 32]) * 0.125f;
      float mx = fmaxf(r0, r1);
#pragma unroll
      for (int msk = 16; msk >= 1; msk >>= 1)
        mx = fmaxf(mx, __shfl_xor(mx, msk, 32));
      const float e0 = __expf(r0 - mx);
      const float e1 = __expf(r1 - mx);
      float sm = e0 + e1;
#pragma unroll
      for (int msk = 16; msk >= 1; msk >>= 1)
        sm += __shfl_xor(sm, msk, 32);
      const float qw = qj / sm;
      o0 += qw * e0;
      o1 += qw * e1;
    }
    os[p * QS + lane]      = o0;
    os[p * QS + lane + 32] = o1;
  }
  __syncthreads();

  // ---- phase 4: out @ Wout + bout  (16x64)@(64x128) via WMMA ----
  {
    const int m    = lane & 15;
    const int koff = (lane >> 4) * 2;
    for (int ct = wave; ct < 8; ct += 4) {
      const int n = ct * 16 + (lane & 15);
      v8f acc = {0.f, 0.f, 0.f, 0.f, 0.f, 0.f, 0.f, 0.f};
#pragma unroll 4
      for (int k = 0; k < NC; k += 4) {
        v2f af, bf;
        af.x = os[m * QS + k + koff];
        af.y = os[m * QS + k + koff + 1];
        bf.x = Wout[(k + koff) * Cc + n];
        bf.y = Wout[(k + koff + 1) * Cc + n];
        acc = wmma_k4(af, bf, acc);
      }
      const float bn = bout[n];
#pragma unroll
      for (int i = 0; i < 8; ++i)
        outs[(i + ((lane >> 4) << 3)) * XS + n] = acc[i] + bn;
    }
  }
  __syncthreads();

  // ---- phase 5: residual + LayerNorm (8 lanes per row, shfl reduction) ----
  {
    const int row   = tid >> 3;        // 0..15
    const int cbase = (tid & 7) * 16;  // 8 lanes x 16 channels = 128
    float vals[16];
    float sum = 0.f, sq = 0.f;
#pragma unroll
    for (int i = 0; i < 16; ++i) {
      const float v = xs[row * XS + cbase + i] + outs[row * XS + cbase + i];
      vals[i] = v;
      sum += v;
      sq  += v * v;
    }
#pragma unroll
    for (int msk = 4; msk >= 1; msk >>= 1) {
      sum += __shfl_xor(sum, msk, 32);
      sq  += __shfl_xor(sq,  msk, 32);
    }
    const float mu   = sum * (1.f / 128.f);
    const float var  = sq * (1.f / 128.f) - mu * mu;
    const float rstd = rsqrtf(var + 1e-6f);
#pragma unroll
    for (int i = 0; i < 16; ++i) {
      const int c = cbase + i;
      y[(size_t)(p0 + row) * Cc + c] =
          (vals[i] - mu) * rstd * ln_scale[c] + ln_bias[c];
    }
  }
}

// ---------------------------------------------------------------------------
// Host launcher
// ---------------------------------------------------------------------------
extern "C" void kernel_launch(void* const* d_in, const int* in_sizes, int n_in,
                              void* d_out, int out_size, void* d_ws, size_t ws_size,
                              hipStream_t stream) {
  const float* x     = (const float*)d_in[0];
  const float* s     = (const float*)d_in[1];
  const float* amask = (const float*)d_in[2];
  const float* Wq    = (const float*)d_in[3];
  const float* bq    = (const float*)d_in[4];
  const float* Wkx   = (const float*)d_in[5];
  const float* bkx   = (const float*)d_in[6];
  const float* Wvx   = (const float*)d_in[7];
  const float* bvx   = (const float*)d_in[8];
  const float* Wks   = (const float*)d_in[9];
  const float* bks   = (const float*)d_in[10];
  const float* Wvs   = (const float*)d_in[11];
  const float* bvs   = (const float*)d_in[12];
  const float* Wout  = (const float*)d_in[13];
  const float* bout  = (const float*)d_in[14];
  const float* lns   = (const float*)d_in[15];
  const float* lnb   = (const float*)d_in[16];
  float* y = (float*)d_out;

  // workspace layout (all f32): ks, vs, G, Ksum, Vsum  (~578 KB total)
  char* ws = (char*)d_ws;
  float* ksb  = (float*)(ws + 0);                       // B*S*NC
  float* vsb  = (float*)(ws + 262144);                  // B*S*NC
  float* G    = (float*)(ws + 524288);                  // B*NC*NC
  float* Ksum = (float*)(ws + 589824);                  // B*NC
  float* Vsum = (float*)(ws + 590848);                  // B*NC

  text_proj_kernel<<<dim3(Bn * (Ss / 16) * 2), dim3(128), 0, stream>>>(
      s, amask, Wks, bks, Wvs, bvs, ksb, vsb);
  colsum_kernel<<<dim3(Bn), dim3(128), 0, stream>>>(ksb, vsb, Ksum, Vsum);
  gram_kernel<<<dim3(Bn * (NC / 16)), dim3(128), 0, stream>>>(ksb, vsb, G);
  fused_attn_kernel<<<dim3(NPOS / 16), dim3(128), 0, stream>>>(
      x, Wq, bq, Wkx, bkx, Wvx, bvx, G, Ksum, Vsum, Wout, bout, lns, lnb, y);
}